// CustomCombinedLoss_82635170775477
// MI455X (gfx1250) — compile-verified
//
#include <hip/hip_runtime.h>

#define LOSS_WEIGHT 2.0f
#define LOSS_MARGIN 0.5f
#define B_DIM 16384
#define C_DIM 2048

typedef float v2f __attribute__((ext_vector_type(2)));
typedef float v4f __attribute__((ext_vector_type(4)));
typedef float v8f __attribute__((ext_vector_type(8)));

// Wave32 all-lane sum. Uses V_WMMA_F32_16X16X4_F32 with B = ones:
// lane L supplies A component {p_L, 0}; per the 16x4 A layout (VGPR0 lanes0-15 K=0,
// lanes16-31 K=2; VGPR1 K=1/K=3) the A row-sum for row m is p_m + p_{16+m}.
// With B all ones, D[m][n] = p_m + p_{16+m} for every n. Lane n (n<16) holds
// D[0..7][n] in its 8 D VGPRs, lane 16+n holds D[8..15][n]; summing the 8 regs
// and adding the lane^16 partner yields the full 32-lane sum in every lane.
__device__ __forceinline__ float wave_sum32(float p) {
#if __has_builtin(__builtin_amdgcn_wmma_f32_16x16x4_f32)
  v2f a; a[0] = p;    a[1] = 0.0f;
  v2f b; b[0] = 1.0f; b[1] = 1.0f;
  v8f c = {0.0f, 0.0f, 0.0f, 0.0f, 0.0f, 0.0f, 0.0f, 0.0f};
  v8f d = __builtin_amdgcn_wmma_f32_16x16x4_f32(
      /*neg_a=*/false, a, /*neg_b=*/false, b,
      /*c_mod=*/(short)0, c, /*reuse_a=*/false, /*reuse_b=*/false);
  float s = ((d[0] + d[1]) + (d[2] + d[3])) + ((d[4] + d[5]) + (d[6] + d[7]));
  s += __shfl_xor(s, 16, 32);
  return s;
#else
  float s = p;
  s += __shfl_xor(s, 1, 32);
  s += __shfl_xor(s, 2, 32);
  s += __shfl_xor(s, 4, 32);
  s += __shfl_xor(s, 8, 32);
  s += __shfl_xor(s, 16, 32);
  return s;
#endif
}

// One block (256 threads = 8 wave32) per row. Streams the 8 KiB row with
// non-temporal b128 loads, computes the hinge sum, reduces it, and writes
// the per-row contribution (bce_i + mml_i) to the workspace.
__global__ __launch_bounds__(256) void row_loss_kernel(
    const float* __restrict__ positive_prob,
    const float* __restrict__ predictions,
    const int*   __restrict__ target,
    float*       __restrict__ row_partial) {
  const int row = blockIdx.x;
  const int tid = threadIdx.x;
  const int t   = target[row];   // uniform across block
  const int tp  = t - 1;         // -1 => negative sample, skip margin loss

  float rs = 0.0f;
  if (tp >= 0) {                 // block-uniform branch (EXEC stays all-ones)
    const float xy   = predictions[(size_t)row * C_DIM + tp];  // uniform load
    const float base = LOSS_MARGIN - xy;
    const v4f* rowp  = (const v4f*)(predictions + (size_t)row * C_DIM);
    // 256 threads * 2 * float4 = 2048 floats = full row, coalesced b128 loads
    v4f q0 = __builtin_nontemporal_load(rowp + tid);
    v4f q1 = __builtin_nontemporal_load(rowp + 256 + tid);
    float r0 = (fmaxf(0.0f, base + q0[0]) + fmaxf(0.0f, base + q0[1])) +
               (fmaxf(0.0f, base + q0[2]) + fmaxf(0.0f, base + q0[3]));
    float r1 = (fmaxf(0.0f, base + q1[0]) + fmaxf(0.0f, base + q1[1])) +
               (fmaxf(0.0f, base + q1[2]) + fmaxf(0.0f, base + q1[3]));
    rs = r0 + r1;
  }

  const float wsum = wave_sum32(rs);   // full EXEC: runs for all rows

  __shared__ float wpart[8];
  if ((tid & 31) == 0) wpart[tid >> 5] = wsum;
  __syncthreads();

  if (tid == 0) {
    float rowsum = 0.0f;
    #pragma unroll
    for (int i = 0; i < 8; ++i) rowsum += wpart[i];   // fixed order: deterministic
    // j==y term of the unmasked sum is exactly max(0, MARGIN) = MARGIN; remove it.
    const float mml = (tp >= 0) ? (rowsum - LOSS_MARGIN) * (1.0f / (float)C_DIM)
                                : 0.0f;
    // weighted BCE term, torch-style log clamp at -100
    const float p     = positive_prob[row];
    const float pos_t = (t != 0) ? 1.0f : 0.0f;
    const float w     = (pos_t < 1.0f) ? LOSS_WEIGHT : 1.0f;
    const float lp    = fmaxf(logf(p),     -100.0f);
    const float l1p   = fmaxf(log1pf(-p),  -100.0f);
    const float bce   = -(w * (pos_t * lp + (1.0f - pos_t) * l1p));
    row_partial[row] = bce + mml;
  }
}

// Single-block deterministic final reduction of the 16384 row partials.
__global__ __launch_bounds__(256) void final_reduce_kernel(
    const float* __restrict__ row_partial,
    float*       __restrict__ out) {
  const int tid = threadIdx.x;
  float s = 0.0f;
  #pragma unroll 8
  for (int k = 0; k < B_DIM / 256; ++k)    // fixed per-thread order, coalesced
    s += row_partial[tid + 256 * k];

  const float wsum = wave_sum32(s);

  __shared__ float wpart[8];
  if ((tid & 31) == 0) wpart[tid >> 5] = wsum;
  __syncthreads();

  if (tid == 0) {
    float tot = 0.0f;
    #pragma unroll
    for (int i = 0; i < 8; ++i) tot += wpart[i];
    out[0] = tot;
  }
}

extern "C" void kernel_launch(void* const* d_in, const int* in_sizes, int n_in,
                              void* d_out, int out_size, void* d_ws, size_t ws_size,
                              hipStream_t stream) {
  (void)in_sizes; (void)n_in; (void)out_size; (void)ws_size;
  const float* positive_prob = (const float*)d_in[0];
  const float* predictions   = (const float*)d_in[1];
  const int*   target        = (const int*)d_in[2];
  float* out         = (float*)d_out;
  float* row_partial = (float*)d_ws;   // needs B_DIM * 4 = 64 KiB of scratch

  row_loss_kernel<<<B_DIM, 256, 0, stream>>>(positive_prob, predictions, target,
                                             row_partial);
  final_reduce_kernel<<<1, 256, 0, stream>>>(row_partial, out);
}